// CrossAttention_47502338294587
// MI455X (gfx1250) — compile-verified
//
#include <hip/hip_runtime.h>

typedef __attribute__((ext_vector_type(2))) float v2f;
typedef __attribute__((ext_vector_type(8))) float v8f;

#define C_DIM 1024
#define B_DIM 4

// ---------------------------------------------------------------------------
// GEMV via fp32 WMMA:  out[r][n] = sum_k A[r][k] * W[k][n] + bias[n], r=0..3
// A is [4][1024] (padded to 16 rows with zeros via lane mask).
// Grid: 64 blocks (one 16-wide column tile each) x 128 threads (4 waves).
// Each wave covers K/4 = 256 of the K dimension -> 64 chained V_WMMA_F32_16X16X4_F32.
// Partial 16x16 accumulators reduced across waves with LDS float atomics.
// ---------------------------------------------------------------------------
__global__ __launch_bounds__(128)
void gemv_wmma_f32_kernel(const float* __restrict__ A,    // [4][C_DIM]
                          const float* __restrict__ W,    // [C_DIM][C_DIM]
                          const float* __restrict__ bias, // [C_DIM]
                          float* __restrict__ out)        // [4][C_DIM]
{
    __shared__ float red[B_DIM][16];

    const int tid  = threadIdx.x;
    const int lane = tid & 31;
    const int wave = tid >> 5;           // 0..3 : K-split
    const int n0   = blockIdx.x << 4;    // column tile base
    const int row  = lane & 15;          // matrix row (A) / col-in-tile (B)
    const int hi   = lane >> 4;          // lane group 0/1 -> K sub-offset

    // A fragment addressing: rows >= B_DIM contribute zero (branchless mask,
    // EXEC stays all-ones for the WMMA as the ISA requires).
    const int   ar = (row < B_DIM) ? row : 0;
    const float am = (row < B_DIM) ? 1.0f : 0.0f;
    const int   col = n0 + row;          // this lane's B/D column

    if (tid < B_DIM * 16) red[tid >> 4][tid & 15] = 0.0f;
    __syncthreads();

    v8f acc = {};
    const int kbeg = wave * (C_DIM / 4);
    const int kend = kbeg + (C_DIM / 4);
    #pragma unroll 4
    for (int k = kbeg; k < kend; k += 4) {
        const int ka = k + 2 * hi;       // K pair handled by this lane group
        v2f a, b;
        // A 16x4 fragment (f32): lane holds A[row][ka], A[row][ka+1]
        a.x = A[ar * C_DIM + ka    ] * am;
        a.y = A[ar * C_DIM + ka + 1] * am;
        // B 4x16 fragment (f32): lane holds W[ka][col], W[ka+1][col]
        b.x = W[(ka    ) * C_DIM + col];
        b.y = W[(ka + 1) * C_DIM + col];
        // (neg_a, A, neg_b, B, c_mod, C, reuse_a, reuse_b)
        acc = __builtin_amdgcn_wmma_f32_16x16x4_f32(
                  false, a, false, b, (short)0, acc, false, false);
    }

    // D layout: VGPR r, lanes 0-15 => D[M=r][N=lane]. Only rows 0..3 needed.
    if (hi == 0) {
        #pragma unroll
        for (int r = 0; r < B_DIM; ++r)
            atomicAdd(&red[r][row], acc[r]);   // ds_add_f32, cross-wave K reduce
    }
    __syncthreads();

    if (tid < 16) {
        const int c = n0 + tid;
        const float bb = bias[c];
        #pragma unroll
        for (int r = 0; r < B_DIM; ++r)
            out[r * C_DIM + c] = red[r][tid] + bb;
    }
}

// ---------------------------------------------------------------------------
// Broadcast:  out[b][t][:] = rows[b][:]   (the 16 MB streaming-store stage)
// One block per (b,t) row: 256 threads x float4 = 1024 floats.
// rows[] is 16 KB and stays resident in cache across all 4096 blocks.
// ---------------------------------------------------------------------------
__global__ __launch_bounds__(256)
void broadcast_rows_kernel(const float* __restrict__ rows, // [4][C_DIM]
                           float* __restrict__ out)        // [4][T][C_DIM]
{
    const int bt = blockIdx.x;          // b*T + t
    const int b  = bt >> 10;            // T = 1024
    const float4 v = ((const float4*)(rows + b * C_DIM))[threadIdx.x];
    ((float4*)(out + (size_t)bt * C_DIM))[threadIdx.x] = v;
}

// ---------------------------------------------------------------------------
// Inputs (setup_inputs order):
//  0:x  1:visual_features  2:Wq 3:bq 4:Wk 5:bk 6:Wv 7:bv 8:Wp 9:bp
// Math: out[b,t,:] = (vf[b,:] @ Wv + bv) @ Wp + bp   (x/Wq/Wk unused — exact)
// ---------------------------------------------------------------------------
extern "C" void kernel_launch(void* const* d_in, const int* in_sizes, int n_in,
                              void* d_out, int out_size, void* d_ws, size_t ws_size,
                              hipStream_t stream) {
    const float* vf = (const float*)d_in[1];
    const float* Wv = (const float*)d_in[6];
    const float* bv = (const float*)d_in[7];
    const float* Wp = (const float*)d_in[8];
    const float* bp = (const float*)d_in[9];
    float* out = (float*)d_out;

    float* vv   = (float*)d_ws;          // [4][1024] = 16 KB
    float* orow = vv + B_DIM * C_DIM;    // [4][1024] = 16 KB

    // vv = vf @ Wv + bv
    gemv_wmma_f32_kernel<<<C_DIM / 16, 128, 0, stream>>>(vf, Wv, bv, vv);
    // orow = vv @ Wp + bp
    gemv_wmma_f32_kernel<<<C_DIM / 16, 128, 0, stream>>>(vv, Wp, bp, orow);
    // out[b,t,:] = orow[b,:]
    broadcast_rows_kernel<<<B_DIM * 1024, 256, 0, stream>>>(orow, out);
}